// GCN_841813590026
// MI455X (gfx1250) — compile-verified
//
#include <hip/hip_runtime.h>
#include <stdint.h>

typedef __attribute__((ext_vector_type(2))) float v2f;
typedef __attribute__((ext_vector_type(8))) float v8f;
typedef __attribute__((ext_vector_type(4))) int   v4i;

#if defined(__has_builtin)
#  if __has_builtin(__builtin_amdgcn_global_load_async_to_lds_b128)
#    define USE_ASYNC_LDS 1
#  else
#    define USE_ASYNC_LDS 0
#  endif
#else
#  define USE_ASYNC_LDS 0
#endif

#if USE_ASYNC_LDS
typedef __attribute__((address_space(1))) v4i* as1_v4i_p;  // 'int4 __device__ *'
typedef __attribute__((address_space(3))) v4i* as3_v4i_p;  // 'int4 __shared__ *'
#endif

// ---------------------------------------------------------------------------
// Degree / symmetric-normalization pass
// ---------------------------------------------------------------------------
__global__ void gcn_deg_init(float* __restrict__ deg, int n) {
    int i = blockIdx.x * blockDim.x + threadIdx.x;
    if (i < n) deg[i] = 1.0f;               // +1 for the self-loop
}

__global__ void gcn_deg_count(const int* __restrict__ dst,
                              float* __restrict__ deg, int e) {
    int i = blockIdx.x * blockDim.x + threadIdx.x;
    if (i < e) atomicAdd(&deg[dst[i]], 1.0f);
}

__global__ void gcn_deg_finalize(float* __restrict__ deg, int n) {
    int i = blockIdx.x * blockDim.x + threadIdx.x;
    if (i < n) deg[i] = rsqrtf(deg[i]);     // deg -> deg^{-1/2}
}

// ---------------------------------------------------------------------------
// Dense GEMM  C[M,Ncol] = A[M,K] @ W[K,Ncol]  via V_WMMA_F32_16X16X4_F32.
//
// Block = 256 threads = 8 waves; block tile = 128 rows x 64 cols.
// W slab (32 K-rows x 64 cols) staged in LDS once per block via
// GLOBAL_LOAD_ASYNC_TO_LDS_B128 (ASYNCcnt), double-buffered, padded to
// stride 72 floats so the two 16-lane halves read disjoint bank sets.
// A fragments for a whole slab are preloaded into registers so the
// global_load_b64s clause up and their latency spans 32 WMMAs.
//
// Fragment layout (ISA 7.12.2, 32-bit operands, wave32):
//   A 16x4 : lane L (idx=L&15, half=L>>4) holds A[idx][half*2 + j], j=0,1
//   B 4x16 : lane L holds W[half*2 + j][col=idx]
//   C 16x16: vgpr v of lane L holds C[half*8 + v][idx]
// ---------------------------------------------------------------------------
#define KSLAB   32
#define LDSTR   72            // padded row stride (floats); 72*4 % 16 == 0

__device__ __forceinline__ void gcn_stage_slab(const float* __restrict__ Wg,
                                               float* lds_dst,
                                               int k0, int Ncol, int colBase,
                                               int tid)
{
    // 32 rows x 64 cols = 512 float4 slots; 2 per thread, coalesced by row.
    #pragma unroll
    for (int q = 0; q < 2; ++q) {
        int p  = tid + q * 256;            // 0..511
        int r  = p >> 4;                   // 0..31
        int c4 = (p & 15) << 2;            // 0,4,..,60
        const float* g = Wg + (size_t)(k0 + r) * Ncol + colBase + c4;
        float* l = lds_dst + r * LDSTR + c4;
#if USE_ASYNC_LDS
        __builtin_amdgcn_global_load_async_to_lds_b128(
            (as1_v4i_p)(uintptr_t)g,
            (as3_v4i_p)(unsigned)(uintptr_t)l,   // low 32 bits == LDS offset
            /*offset=*/0, /*cpol=*/0);
#else
        *(float4*)l = *(const float4*)g;
#endif
    }
}

__device__ __forceinline__ void gcn_slab_wait() {
#if USE_ASYNC_LDS
#  if defined(__has_builtin) && __has_builtin(__builtin_amdgcn_s_wait_asynccnt)
    __builtin_amdgcn_s_wait_asynccnt(0);
#  else
    asm volatile("s_wait_asynccnt 0" ::: "memory");
#  endif
#endif
    __syncthreads();
}

__global__ __launch_bounds__(256) void gcn_gemm_wmma_f32(
    const float* __restrict__ A,   // [M, K]
    const float* __restrict__ W,   // [K, Ncol]
    float* __restrict__ C,         // [M, Ncol]
    int M, int K, int Ncol)
{
    __shared__ float Bs[2][KSLAB * LDSTR];     // 2 x 9 KB

    const int tid  = threadIdx.x;
    const int lane = tid & 31;
    const int wave = tid >> 5;                 // 0..7
    const int half = lane >> 4;                // 0 | 1
    const int idx  = lane & 15;

    const int rowTile = blockIdx.x * 8 + wave; // 16-row tile index
    const int colBase = blockIdx.y * 64;       // 4 x 16-col tiles
    const bool active = (rowTile * 16 < M);    // wave-uniform; EXEC stays full

    const float* __restrict__ arow =
        A + (size_t)(rowTile * 16 + idx) * K;  // address only used if active

    v8f acc0 = {}, acc1 = {}, acc2 = {}, acc3 = {};

    // Prologue: stage slab 0 (all waves participate — slab is block-shared).
    gcn_stage_slab(W, &Bs[0][0], 0, Ncol, colBase, tid);
    gcn_slab_wait();

    const int nslab = K / KSLAB;
    for (int s = 0; s < nslab; ++s) {
        const int kb = s * KSLAB;

        // Issue next slab into the other buffer while computing this one.
        if (s + 1 < nslab)
            gcn_stage_slab(W, &Bs[(s + 1) & 1][0], kb + KSLAB, Ncol, colBase, tid);

        if (active) {
            if (kb + 128 < K)
                __builtin_prefetch(arow + kb + 128, 0, 1);  // stream A rows

            // Preload the slab's 8 A fragments -> one clause of b64 loads.
            v2f af[KSLAB / 4];
            #pragma unroll
            for (int j = 0; j < KSLAB / 4; ++j) {
                const float* ap = arow + kb + j * 4 + half * 2;
                af[j].x = ap[0];
                af[j].y = ap[1];
            }

            const float* Bl = &Bs[s & 1][0];
            #pragma unroll
            for (int kk = 0; kk < KSLAB; kk += 4) {
                const v2f afrag = af[kk >> 2];
                const float* l0 = Bl + (kk + half * 2) * LDSTR + idx;
                const float* l1 = l0 + LDSTR;
                v2f b;
                b.x = l0[0];  b.y = l1[0];
                acc0 = __builtin_amdgcn_wmma_f32_16x16x4_f32(false, afrag, false, b,
                                                             (short)0, acc0, false, false);
                b.x = l0[16]; b.y = l1[16];
                acc1 = __builtin_amdgcn_wmma_f32_16x16x4_f32(false, afrag, false, b,
                                                             (short)0, acc1, false, false);
                b.x = l0[32]; b.y = l1[32];
                acc2 = __builtin_amdgcn_wmma_f32_16x16x4_f32(false, afrag, false, b,
                                                             (short)0, acc2, false, false);
                b.x = l0[48]; b.y = l1[48];
                acc3 = __builtin_amdgcn_wmma_f32_16x16x4_f32(false, afrag, false, b,
                                                             (short)0, acc3, false, false);
            }
        }

        if (s + 1 < nslab)
            gcn_slab_wait();   // copies done + all waves finished reading
    }

    if (active) {
        // vgpr v of lane L -> C[rowTile*16 + half*8 + v][colBase + t*16 + idx]
        const int rBase = rowTile * 16 + half * 8;
        #pragma unroll
        for (int v = 0; v < 8; ++v) {
            float* crow = C + (size_t)(rBase + v) * Ncol + colBase + idx;
            crow[0]  = acc0[v];
            crow[16] = acc1[v];
            crow[32] = acc2[v];
            crow[48] = acc3[v];
        }
    }
}

// ---------------------------------------------------------------------------
// out[i][c] = h[i][c] * dinv[i]^2 + bias[c]   (self-loop + bias, pre-scatter)
// ---------------------------------------------------------------------------
__global__ void gcn_agg_init(const float* __restrict__ h,
                             const float* __restrict__ dinv,
                             const float* __restrict__ bias,
                             float* __restrict__ out,
                             int n, int Cch)
{
    long long gid = (long long)blockIdx.x * blockDim.x + threadIdx.x;
    long long total = (long long)n * Cch;
    if (gid >= total) return;
    int i = (int)(gid / Cch);
    int c = (int)(gid % Cch);
    float di = dinv[i];
    out[gid] = h[gid] * di * di + bias[c];
}

// ---------------------------------------------------------------------------
// Edge scatter: out[dst] += h[src] * dinv[src]*dinv[dst]
// Cch/4 threads per edge -> float4 gathers are fully coalesced, and all lanes
// of a wave work on the same edge so src/dst/coef are wave-uniform.
// ---------------------------------------------------------------------------
__global__ void gcn_agg_edges(const float* __restrict__ h,
                              const int* __restrict__ src,
                              const int* __restrict__ dst,
                              const float* __restrict__ dinv,
                              float* __restrict__ out,
                              int nEdges, int Cch)
{
    const int tpe = Cch >> 2;                     // threads per edge
    long long gid = (long long)blockIdx.x * blockDim.x + threadIdx.x;
    long long total = (long long)nEdges * tpe;
    if (gid >= total) return;
    int e  = (int)(gid / tpe);
    int c  = (int)(gid % tpe) << 2;
    int s  = src[e];
    int d  = dst[e];
    float coef = dinv[s] * dinv[d];
    const float4 hv = *(const float4*)(h + (size_t)s * Cch + c);
    float* o = out + (size_t)d * Cch + c;
    atomicAdd(o + 0, hv.x * coef);
    atomicAdd(o + 1, hv.y * coef);
    atomicAdd(o + 2, hv.z * coef);
    atomicAdd(o + 3, hv.w * coef);
}

// ---------------------------------------------------------------------------
// jax.nn.gelu default = tanh approximation
// ---------------------------------------------------------------------------
__global__ void gcn_gelu(float* __restrict__ buf, long long n) {
    long long gid = (long long)blockIdx.x * blockDim.x + threadIdx.x;
    if (gid >= n) return;
    float x = buf[gid];
    float inner = 0.7978845608028654f * (x + 0.044715f * x * x * x);
    buf[gid] = 0.5f * x * (1.0f + tanhf(inner));
}

// ---------------------------------------------------------------------------
// Host-side orchestration
// ---------------------------------------------------------------------------
extern "C" void kernel_launch(void* const* d_in, const int* in_sizes, int n_in,
                              void* d_out, int out_size, void* d_ws, size_t ws_size,
                              hipStream_t stream) {
    (void)n_in; (void)out_size; (void)ws_size;

    const int IN = 512, HID = 512, OUT = 256;

    const float* x   = (const float*)d_in[0];
    const int*   ei  = (const int*)  d_in[1];
    const float* W1  = (const float*)d_in[2];
    const float* b1  = (const float*)d_in[3];
    const float* W2  = (const float*)d_in[4];
    const float* b2  = (const float*)d_in[5];
    const float* W3  = (const float*)d_in[6];
    const float* b3  = (const float*)d_in[7];
    float* out = (float*)d_out;

    const int N = in_sizes[0] / IN;       // 50000
    const int E = in_sizes[1] / 2;        // 800000
    const int* src = ei;                  // edge_index[0]
    const int* dst = ei + E;              // edge_index[1]

    // Workspace: dinv[N] | bufA[N*HID] | bufB[N*HID]
    char* wsb = (char*)d_ws;
    float* dinv = (float*)wsb;
    size_t off   = (((size_t)N * sizeof(float)) + 255) & ~(size_t)255;
    size_t hbyte = (((size_t)N * HID * sizeof(float)) + 255) & ~(size_t)255;
    float* bufA = (float*)(wsb + off);
    float* bufB = (float*)(wsb + off + hbyte);

    const int T = 256;
    auto cdiv = [](long long a, long long b) { return (unsigned)((a + b - 1) / b); };

    // --- degree / dinv ---
    gcn_deg_init    <<<cdiv(N, T), T, 0, stream>>>(dinv, N);
    gcn_deg_count   <<<cdiv(E, T), T, 0, stream>>>(dst, dinv, E);
    gcn_deg_finalize<<<cdiv(N, T), T, 0, stream>>>(dinv, N);

    const unsigned gRows = cdiv(N, 128);  // 8 waves x 16 rows per block

    // --- layer 1: x(IN) -> bufB(HID) ---
    gcn_gemm_wmma_f32<<<dim3(gRows, HID / 64), T, 0, stream>>>(x, W1, bufA, N, IN, HID);
    gcn_agg_init <<<cdiv((long long)N * HID, T), T, 0, stream>>>(bufA, dinv, b1, bufB, N, HID);
    gcn_agg_edges<<<cdiv((long long)E * (HID / 4), T), T, 0, stream>>>(bufA, src, dst, dinv, bufB, E, HID);
    gcn_gelu     <<<cdiv((long long)N * HID, T), T, 0, stream>>>(bufB, (long long)N * HID);

    // --- layer 2: bufB(HID) -> bufB(HID) ---
    gcn_gemm_wmma_f32<<<dim3(gRows, HID / 64), T, 0, stream>>>(bufB, W2, bufA, N, HID, HID);
    gcn_agg_init <<<cdiv((long long)N * HID, T), T, 0, stream>>>(bufA, dinv, b2, bufB, N, HID);
    gcn_agg_edges<<<cdiv((long long)E * (HID / 4), T), T, 0, stream>>>(bufA, src, dst, dinv, bufB, E, HID);
    gcn_gelu     <<<cdiv((long long)N * HID, T), T, 0, stream>>>(bufB, (long long)N * HID);

    // --- layer 3: bufB(HID) -> out(OUT) ---
    gcn_gemm_wmma_f32<<<dim3(gRows, OUT / 64), T, 0, stream>>>(bufB, W3, bufA, N, HID, OUT);
    gcn_agg_init <<<cdiv((long long)N * OUT, T), T, 0, stream>>>(bufA, dinv, b3, out, N, OUT);
    gcn_agg_edges<<<cdiv((long long)E * (OUT / 4), T), T, 0, stream>>>(bufA, src, dst, dinv, out, E, OUT);
}